// ConsensusAttention_20796231647906
// MI455X (gfx1250) — compile-verified
//
#include <hip/hip_runtime.h>
#include <float.h>

typedef __attribute__((ext_vector_type(2))) float v2f;
typedef __attribute__((ext_vector_type(8))) float v8f;

#define BB   8
#define NN   1024
#define LL   6
#define DD   512
#define GS   32            // 32x32 patch grid
#define NKEY 160           // 5 grid rows * 32 keys
#define SLDP 164           // padded LDS row stride (floats)
#define PSTRIDE (LL*DD)    // 3072 floats per patch
#define INV_SQRT_D 0.044194173824159216f  // 1/sqrt(512)

// ---------- global sum-of-squares reduction (deterministic 2-stage) ----------
__global__ void ssq_partial_kernel(const float* __restrict__ x,
                                   float* __restrict__ partial, long long n) {
  __shared__ float red[256];
  long long stride = (long long)gridDim.x * blockDim.x;
  float s = 0.f;
  for (long long i = (long long)blockIdx.x * blockDim.x + threadIdx.x; i < n; i += stride) {
    float v = x[i];
    s += v * v;
  }
  red[threadIdx.x] = s;
  __syncthreads();
  for (int off = 128; off > 0; off >>= 1) {
    if ((int)threadIdx.x < off) red[threadIdx.x] += red[threadIdx.x + off];
    __syncthreads();
  }
  if (threadIdx.x == 0) partial[blockIdx.x] = red[0];
}

__global__ void ssq_final_kernel(const float* __restrict__ partial,
                                 float* __restrict__ scale_out, int np) {
  __shared__ float red[256];
  float s = 0.f;
  for (int i = threadIdx.x; i < np; i += 256) s += partial[i];
  red[threadIdx.x] = s;
  __syncthreads();
  for (int off = 128; off > 0; off >>= 1) {
    if ((int)threadIdx.x < off) red[threadIdx.x] += red[threadIdx.x + off];
    __syncthreads();
  }
  if (threadIdx.x == 0) {
    float ssq = fmaxf(red[0], 1e-12f);
    scale_out[0] = rsqrtf(ssq) * INV_SQRT_D;
  }
}

// ---------- main: sparse local attention with fp32 WMMA ----------
__global__ __launch_bounds__(256) void consensus_attn_kernel(
    const float* __restrict__ levels, float* __restrict__ out,
    const float* __restrict__ scale_ptr) {
  __shared__ float S[32 * SLDP];   // scores, then softmax probs P

  const int bid = blockIdx.x;
  const int r  = bid & 31;              // query grid row
  const int l  = (bid >> 5) % LL;
  const int bb = bid / (32 * LL);
  const float scale = scale_ptr[0];

  const int tid  = threadIdx.x;
  const int wave = tid >> 5;
  const int lane = tid & 31;
  const int half = lane >> 4;           // 0: lanes 0-15, 1: lanes 16-31
  const int mrow = lane & 15;

  // pointer to (bb, patch 0, l, feature 0); patch i is at +i*PSTRIDE
  const float* base = levels + (size_t)bb * NN * PSTRIDE + (size_t)l * DD;

  // ======== Phase A: S = scale * Q K^T, masked ========
  // 20 tiles: mt in {0,1} (query halves), nt in 0..9 (16-key tiles)
  for (int tile = wave; tile < 20; tile += 8) {
    const int mt = tile & 1;
    const int nt = tile >> 1;
    const int ky = r - 2 + (nt >> 1);          // uniform per tile
    const int dy = (nt >> 1) - 2;
    const int kx = ((nt & 1) << 4) + mrow;     // this lane's key grid-x (N role)

    if (ky < 0 || ky >= GS) {                  // off-grid key row -> fully masked
      #pragma unroll
      for (int v = 0; v < 8; ++v) {
        const int M = v + (half << 3);
        S[((mt << 4) + M) * SLDP + (nt << 4) + mrow] = -FLT_MAX;
      }
      continue;
    }

    const int qi = (r << 5) + (mt << 4) + mrow;  // query patch (A role, M=mrow)
    const int ki = (ky << 5) + kx;               // key patch   (B role, N=mrow)
    const float* qp = base + (size_t)qi * PSTRIDE + (half << 1);
    const float* kp = base + (size_t)ki * PSTRIDE + (half << 1);

    v8f acc = {};
    #pragma unroll 8
    for (int dd = 0; dd < DD; dd += 4) {
      v2f a = *(const v2f*)(qp + dd);   // A[M=mrow][K = dd+2*half .. +1]
      v2f b = *(const v2f*)(kp + dd);   // B[K][N=mrow] = key feature pair
      acc = __builtin_amdgcn_wmma_f32_16x16x4_f32(false, a, false, b,
                                                  (short)0, acc, false, false);
    }
    #pragma unroll
    for (int v = 0; v < 8; ++v) {
      const int M  = v + (half << 3);
      const int qx = (mt << 4) + M;
      const int dx = kx - qx;
      const int d2 = dx * dx + dy * dy;
      float val = acc[v] * scale;
      if (d2 == 0)      val = -0.0005f;   // ATTEND_SELF=False diagonal value
      else if (d2 > 4)  val = -FLT_MAX;   // dist > radius 2.0
      S[qx * SLDP + (nt << 4) + mrow] = val;
    }
  }
  __syncthreads();

  // ======== Phase B: row softmax (8 threads per row) ========
  {
    const int row = tid >> 3;
    const int sub = tid & 7;
    float* Srow = S + row * SLDP;
    float m = -FLT_MAX;
    for (int k = sub; k < NKEY; k += 8) m = fmaxf(m, Srow[k]);
    for (int off = 4; off > 0; off >>= 1) m = fmaxf(m, __shfl_xor(m, off, 8));
    float ssum = 0.f;
    for (int k = sub; k < NKEY; k += 8) {
      float e = __expf(Srow[k] - m);     // -FLT_MAX entries underflow to 0
      Srow[k] = e;
      ssum += e;
    }
    for (int off = 4; off > 0; off >>= 1) ssum += __shfl_xor(ssum, off, 8);
    const float inv = 1.0f / ssum;
    for (int k = sub; k < NKEY; k += 8) Srow[k] *= inv;
  }
  __syncthreads();

  // ======== Phase C: out = P V ========
  // 64 tiles: mt in {0,1}, nf in 0..31 (feature tiles of 16)
  for (int tile = wave; tile < 64; tile += 8) {
    const int mt = tile & 1;
    const int nf = tile >> 1;
    const int f  = (nf << 4) + mrow;      // this lane's output feature (N role)
    v8f acc = {};
    for (int g = 0; g < 5; ++g) {
      const int ky = r - 2 + g;
      if (ky < 0 || ky >= GS) continue;   // P==0 there; also avoids OOB loads
      const int kbase = g << 5;
      #pragma unroll 4
      for (int k4 = 0; k4 < 32; k4 += 4) {
        const int kk0 = kbase + k4;
        const int k0  = kk0 + (half << 1);         // this lane's K pair start
        v2f a = *(const v2f*)(S + ((mt << 4) + mrow) * SLDP + k0);  // A=P
        const int ki0 = (ky << 5) + (k0 & 31);
        const float* vp = base + (size_t)ki0 * PSTRIDE + f;
        v2f b;
        b.x = vp[0];                               // V[k0  ][f]
        b.y = vp[PSTRIDE];                         // V[k0+1][f]
        acc = __builtin_amdgcn_wmma_f32_16x16x4_f32(false, a, false, b,
                                                    (short)0, acc, false, false);
      }
    }
    #pragma unroll
    for (int v = 0; v < 8; ++v) {
      const int M  = v + (half << 3);
      const int qi = (r << 5) + (mt << 4) + M;
      out[(((size_t)bb * NN + qi) * LL + l) * DD + f] = acc[v];
    }
  }
}

extern "C" void kernel_launch(void* const* d_in, const int* in_sizes, int n_in,
                              void* d_out, int out_size, void* d_ws, size_t ws_size,
                              hipStream_t stream) {
  (void)in_sizes; (void)n_in; (void)out_size; (void)ws_size;
  const float* levels = (const float*)d_in[0];
  float* out   = (float*)d_out;
  float* ws    = (float*)d_ws;
  float* scale = ws;           // ws[0]
  float* partial = ws + 16;    // 1024 floats of partial sums

  const long long total = (long long)BB * NN * LL * DD;
  ssq_partial_kernel<<<1024, 256, 0, stream>>>(levels, partial, total);
  ssq_final_kernel<<<1, 256, 0, stream>>>(partial, scale, 1024);
  consensus_attn_kernel<<<BB * LL * GS, 256, 0, stream>>>(levels, out, scale);
}